// Experts_47579647705876
// MI455X (gfx1250) — compile-verified
//
#include <hip/hip_runtime.h>

typedef __attribute__((ext_vector_type(2))) float v2f;
typedef __attribute__((ext_vector_type(8))) float v8f;
typedef __attribute__((ext_vector_type(4))) int   v4i;

// pointer flavors for the async global->LDS builtin (AS1 = global, AS3 = LDS)
typedef __attribute__((address_space(1))) v4i gv4i;
typedef __attribute__((address_space(3))) v4i lv4i;

#define N_TOT 16384
#define E_TOT 64
#define D_DIM 128
#define LDS_STRIDE 144   // 4*144 ≡ 0 (mod 64 banks); half-waves land 32 banks apart -> conflict-free B loads

#if defined(__HIP_DEVICE_COMPILE__) && __has_builtin(__builtin_amdgcn_global_load_async_to_lds_b128)
#define USE_ASYNC_LDS 1
#else
#define USE_ASYNC_LDS 0
#endif

__device__ __forceinline__ void wait_async_all() {
#if __has_builtin(__builtin_amdgcn_s_wait_asynccnt)
    __builtin_amdgcn_s_wait_asynccnt(0);
#else
    asm volatile("s_wait_asynccnt 0" ::: "memory");
#endif
}

__global__ __launch_bounds__(256) void experts_wmma_f32(
    const float* __restrict__ X,   // [N, D]
    const float* __restrict__ W,   // [E, D, D]  (d-major, f-minor)
    const float* __restrict__ Bb,  // [E, D]
    float* __restrict__ Out)       // [N, E, D]
{
    __shared__ float lw[D_DIM * LDS_STRIDE];   // 73,728 B of 320 KB WGP LDS

    const int e     = blockIdx.x & (E_TOT - 1);
    const int ntile = blockIdx.x >> 6;
    const int n0    = ntile * 128;

    const int tid  = threadIdx.x;
    const int wave = tid >> 5;
    const int lane = tid & 31;
    const int half = lane >> 4;   // 0: lanes 0-15, 1: lanes 16-31
    const int l16  = lane & 15;

    // ---- stage W_e (128x128 fp32) into padded LDS: 2 threads/row, 16x b128 each ----
    {
        const float* wsrc   = W + (size_t)e * (D_DIM * D_DIM);
        const int row       = tid >> 1;
        const int colbase   = (tid & 1) * 64;
        const float* src    = wsrc + row * D_DIM + colbase;
        float*       dst    = lw + row * LDS_STRIDE + colbase;
#if USE_ASYNC_LDS
        // CDNA5 async DMA: global -> LDS without a VGPR bounce, tracked by ASYNCcnt
#pragma unroll
        for (int i = 0; i < 16; ++i) {
            __builtin_amdgcn_global_load_async_to_lds_b128(
                (gv4i*)(uintptr_t)(src + 4 * i),   // global (AS1) v4i*
                (lv4i*)(dst + 4 * i),              // LDS (AS3) v4i*
                0, 0);
        }
        wait_async_all();
#else
        const float4* src4 = (const float4*)src;
#pragma unroll
        for (int i = 0; i < 16; ++i) {
            float4 v = src4[i];
            dst[4*i+0] = v.x; dst[4*i+1] = v.y; dst[4*i+2] = v.z; dst[4*i+3] = v.w;
        }
#endif
    }

    // ---- A fragments for this wave's 16 rows, kept in registers (L2-resident X) ----
    // ISA 32-bit A 16x4 layout: VGPR0 = K={0|2} per half-wave, VGPR1 = K={1|3}
    const int arow = n0 + wave * 16 + l16;
    v2f afrag[32];
    {
        const float* xr = X + (size_t)arow * D_DIM + 2 * half;
#pragma unroll
        for (int s = 0; s < 32; ++s)
            afrag[s] = *(const v2f*)(xr + 4 * s);
    }

    __syncthreads();

    const float* brow  = Bb + e * D_DIM;
    float*       obase = Out + (size_t)(n0 + wave * 16) * (E_TOT * D_DIM) + e * D_DIM;

#pragma unroll
    for (int fp = 0; fp < 4; ++fp) {       // two 16-wide f-tiles per iteration
        const int f0 = fp * 32;
        const int f1 = f0 + 16;

        // bias broadcast: every C VGPR of a lane maps to the same output column
        const float bias0 = brow[f0 + l16];
        const float bias1 = brow[f1 + l16];
        v8f acc0, acc1;
#pragma unroll
        for (int j = 0; j < 8; ++j) { acc0[j] = bias0; acc1[j] = bias1; }

#pragma unroll
        for (int s = 0; s < 32; ++s) {
            const int kb = 4 * s + 2 * half;          // K rows handled by this half-wave
            const float* wp = lw + kb * LDS_STRIDE;
            v2f b0, b1;
            b0.x = wp[f0 + l16];
            b0.y = wp[LDS_STRIDE + f0 + l16];
            b1.x = wp[f1 + l16];
            b1.y = wp[LDS_STRIDE + f1 + l16];
            acc0 = __builtin_amdgcn_wmma_f32_16x16x4_f32(
                       false, afrag[s], false, b0, (short)0, acc0, false, false);
            acc1 = __builtin_amdgcn_wmma_f32_16x16x4_f32(
                       false, afrag[s], false, b1, (short)0, acc1, false, false);
        }

        // C/D layout: VGPR j -> row (j + 8*half), col = lane&15
        // Output is a pure 536 MB write stream: use non-temporal stores (th:NT)
#pragma unroll
        for (int j = 0; j < 8; ++j) {
            const int m = j + half * 8;
            float* orow = obase + (size_t)m * (E_TOT * D_DIM);
            __builtin_nontemporal_store(acc0[j], orow + f0 + l16);
            __builtin_nontemporal_store(acc1[j], orow + f1 + l16);
        }
    }
}

extern "C" void kernel_launch(void* const* d_in, const int* in_sizes, int n_in,
                              void* d_out, int out_size, void* d_ws, size_t ws_size,
                              hipStream_t stream) {
    const float* X  = (const float*)d_in[0];   // (N,1,D) fp32
    const float* W  = (const float*)d_in[1];   // (E,D,D) fp32
    const float* Bb = (const float*)d_in[2];   // (E,D)   fp32
    float* Out      = (float*)d_out;           // (N,E,D) fp32

    dim3 grid((N_TOT / 128) * E_TOT);          // 8192 blocks
    dim3 block(256);                           // 8 waves
    experts_wmma_f32<<<grid, block, 0, stream>>>(X, W, Bb, Out);
}